// ODEFunc_71141838291032
// MI455X (gfx1250) — compile-verified
//
#include <hip/hip_runtime.h>
#include <math.h>

// ---- WMMA vector types (wave32, 16x16 f32 tile = 8 VGPRs) ----
typedef float v2f __attribute__((ext_vector_type(2)));
typedef float v8f __attribute__((ext_vector_type(8)));

#define N_NODES 512
#define HID     8192   // 512*16

// ------------------------------------------------------------
// Async global->LDS copy of 16-byte quanta (CDNA5 ASYNCcnt path).
// dst must be an LDS pointer, src 16B-aligned global.
// ------------------------------------------------------------
__device__ __forceinline__ void async_b128_to_lds(float* ldsDst, const float* gSrc) {
    unsigned loff = (unsigned)(unsigned long long)ldsDst;   // low 32b = LDS offset
    asm volatile("global_load_async_to_lds_b128 %0, %1, off"
                 :: "v"(loff), "v"(gSrc) : "memory");
}
__device__ __forceinline__ void wait_async_all() {
    asm volatile("s_wait_asynccnt 0x0" ::: "memory");
}

// ============================================================
// prep: x0_l1[n][f*8+b] = y[b][n*16+f]   (512 x 128)
// ============================================================
__global__ void prep_x0(const float* __restrict__ y, float* __restrict__ x01) {
    int idx = blockIdx.x * blockDim.x + threadIdx.x;   // 65536
    int b = idx & 7;
    int f = (idx >> 3) & 15;
    int n = idx >> 7;
    x01[idx] = y[(long)b * HID + n * 16 + f];
}

// ============================================================
// Chebyshev GEMM:  Oz = alpha * (Sz @ Xz) [+ beta * Z]
//   block = 256 threads (8 waves); each wave one 16x16 D tile.
//   A: 16 S-rows (contiguous 32KB) async-staged to LDS once.
//   B: X staged in 64-row K-chunks (32KB) -> ds_load B fragments.
// ============================================================
template <int HASZ>
__global__ __launch_bounds__(256) void cheb_mm(
    const float* __restrict__ S, long sStride,
    const float* __restrict__ X, long xStride,
    const float* __restrict__ Z, float beta,
    float* __restrict__ out, long outStride,
    int F, float alpha)
{
    __shared__ float ldsS[16 * 512];                   // 32 KB A stripe
    __shared__ float ldsX[64 * 128];                   // 32 KB B K-chunk
    const int z = blockIdx.z;
    const float* Sz = S + (long)z * sStride;
    const float* Xz = X + (long)z * xStride;
    float* Oz = out + (long)z * outStride;

    const int m0      = blockIdx.x * 16;
    const int colbase = blockIdx.y * 128;
    const int tid     = threadIdx.x;
    const int wave    = tid >> 5;
    const int lane    = tid & 31;
    const int l15     = lane & 15;
    const int kk      = (lane >> 4) << 1;              // 0 or 2 (K split)
    const int n0      = colbase + wave * 16;

    // A stripe: rows m0..m0+15 are one contiguous 8192-float block.
    {
        const float* base = Sz + (long)m0 * 512;
        for (int i = tid; i < 2048; i += 256)          // 2048 x b128
            async_b128_to_lds(ldsS + i * 4, base + i * 4);
    }

    v8f acc = {0.f, 0.f, 0.f, 0.f, 0.f, 0.f, 0.f, 0.f};
    for (int kc = 0; kc < 512; kc += 64) {
        __syncthreads();                               // WAR on ldsX
        for (int i = tid; i < 64 * 32; i += 256) {     // 64 rows x 32 b128
            int row = i >> 5, c4 = (i & 31) << 2;
            async_b128_to_lds(ldsX + row * 128 + c4,
                              Xz + (long)(kc + row) * F + colbase + c4);
        }
        wait_async_all();                              // covers ldsS too (1st iter)
        __syncthreads();

        #pragma unroll
        for (int k = 0; k < 64; k += 4) {
            v2f a, b;
            a.x = ldsS[l15 * 512 + kc + k + kk];       // A[M=l15][K]
            a.y = ldsS[l15 * 512 + kc + k + kk + 1];
            b.x = ldsX[(k + kk) * 128 + wave * 16 + l15];   // B[K][N]
            b.y = ldsX[(k + kk + 1) * 128 + wave * 16 + l15];
            acc = __builtin_amdgcn_wmma_f32_16x16x4_f32(
                false, a, false, b, (short)0, acc, false, false);
        }
    }

    const int rbase = m0 + ((lane >> 4) << 3);         // M = r + 8*(lane>=16)
    const int col   = n0 + l15;
    #pragma unroll
    for (int r = 0; r < 8; ++r) {
        float v = alpha * acc[r];
        if (HASZ) v += beta * Z[(long)(rbase + r) * F + col];
        Oz[(long)(rbase + r) * F + col] = v;
    }
}

// ============================================================
// Hidden dense layer: c = tanh(xs @ W + b), written directly in
// next-layer x0 layout: out[n][o*8+b].  K = fin*M (48 or 272).
// ============================================================
__global__ void dense_hidden(
    const float* __restrict__ x0, const float* __restrict__ T1,
    const float* __restrict__ T2, long supStride, int nSup,
    const float* __restrict__ W, const float* __restrict__ bias,
    float* __restrict__ out, int fin, int fout)
{
    int idx = blockIdx.x * blockDim.x + threadIdx.x;   // n*fout*8 + o*8 + b
    int b = idx & 7;
    int o = (idx >> 3) % fout;
    int n = idx / (fout * 8);
    const int F = fin * 8;
    const int M = 2 * nSup + 1;

    float acc = bias[o];
    for (int m = 0; m < M; ++m) {
        const float* mat;
        if (m == 0) mat = x0;
        else {
            int s = (m - 1) >> 1;
            mat = (m & 1) ? (T1 + (long)s * supStride) : (T2 + (long)s * supStride);
        }
        const float* row  = mat + (long)n * F + b;
        const float* wcol = W + (long)m * fout + o;    // W[(f*M+m)*fout+o]
        for (int f = 0; f < fin; ++f)
            acc = fmaf(row[f * 8], wcol[(long)f * M * fout], acc);
    }
    out[idx] = tanhf(acc);
}

// ============================================================
// Output dense layer: G[rowOff+b][n*16+o] = scale*tanh(xs@W + b)
// (diff rows 0..7 scale=-0.1, adv rows 8..15 scale=-1)
// ============================================================
__global__ void dense_out(
    const float* __restrict__ x0, const float* __restrict__ T1,
    const float* __restrict__ T2, long supStride, int nSup,
    const float* __restrict__ W, const float* __restrict__ bias,
    float* __restrict__ G, int rowOff, float scale)
{
    int idx = blockIdx.x * blockDim.x + threadIdx.x;   // b*8192 + n*16 + o
    int o = idx & 15;
    int n = (idx >> 4) & 511;
    int b = idx >> 13;
    const int fin = 64, F = 512;
    const int fout = 16;
    const int M = 2 * nSup + 1;

    float acc = bias[o];
    for (int m = 0; m < M; ++m) {
        const float* mat;
        if (m == 0) mat = x0;
        else {
            int s = (m - 1) >> 1;
            mat = (m & 1) ? (T1 + (long)s * supStride) : (T2 + (long)s * supStride);
        }
        const float* row  = mat + (long)n * F + b;
        const float* wcol = W + (long)m * fout + o;
        for (int f = 0; f < fin; ++f)
            acc = fmaf(row[f * 8], wcol[(long)f * M * fout], acc);
    }
    G[(long)(rowOff + b) * HID + n * 16 + o] = scale * tanhf(acc);
}

// ============================================================
// Fused gated GEMM:  Y = G(16x8192) @ W_f^T  (single NT-stream
// pass over the 268MB W_f -> HBM-bound floor), gate in-register:
// lane l / l+16 of D VGPR r hold Y_diff[r][col] / Y_adv[r][col].
// ============================================================
__global__ __launch_bounds__(256) void fused_gate_gemm(
    const float* __restrict__ G, const float* __restrict__ Wf,
    const float* __restrict__ bf, float* __restrict__ out)
{
    __shared__ float ldsG[16 * 512];                   // 32 KB A-chunk
    const int tid  = threadIdx.x;
    const int wave = tid >> 5;
    const int lane = tid & 31;
    const int l15  = lane & 15;
    const int kk   = (lane >> 4) << 1;
    const int n0   = (blockIdx.x * 8 + wave) * 16;     // 512 N-tiles total

    const float* wrow = Wf + (long)(n0 + l15) * HID;   // this lane streams its row

    v8f acc = {0.f, 0.f, 0.f, 0.f, 0.f, 0.f, 0.f, 0.f};
    for (int kc = 0; kc < HID; kc += 512) {
        __syncthreads();                               // WAR on ldsG
        for (int i = tid; i < 16 * 128; i += 256) {    // 16 rows x 128 b128
            int row = i >> 7, c4 = (i & 127) << 2;
            async_b128_to_lds(ldsG + row * 512 + c4,
                              G + (long)row * HID + kc + c4);
        }
        wait_async_all();
        __syncthreads();

        __builtin_prefetch(wrow + kc + 512, 0, 1);     // pull next W_f chunk
        for (int k = 0; k < 512; k += 4) {
            v2f a, b;
            a.x = ldsG[l15 * 512 + k + kk];
            a.y = ldsG[l15 * 512 + k + kk + 1];
            // exact per-lane need = 2 consecutive floats of this lane's
            // W_f row; b64 load, non-temporal (one-shot 268MB stream).
            v2f wv = __builtin_nontemporal_load(
                reinterpret_cast<const v2f*>(wrow + kc + k + kk));
            b.x = wv.x;
            b.y = wv.y;
            acc = __builtin_amdgcn_wmma_f32_16x16x4_f32(
                false, a, false, b, (short)0, acc, false, false);
        }
    }

    const int col = n0 + l15;
    const float bias = bf[col];
    #pragma unroll
    for (int r = 0; r < 8; ++r) {
        float mine  = acc[r];
        float other = __shfl(mine, lane ^ 16, 32);
        float Xd = ((lane < 16) ? mine : other) + bias;
        float Xa = ((lane < 16) ? other : mine) + bias;
        float zg  = 1.f / (1.f + expf(-(Xd + Xa)));
        float res = zg * Xd + (1.f - zg) * Xa;
        if (lane < 16) out[(long)r * HID + col] = res;
    }
}

// ============================================================
// Host-side orchestration (all on `stream`, graph-capture safe)
// ============================================================
extern "C" void kernel_launch(void* const* d_in, const int* in_sizes, int n_in,
                              void* d_out, int out_size, void* d_ws, size_t ws_size,
                              hipStream_t stream) {
    const float* y   = (const float*)d_in[1];
    const float* Sd  = (const float*)d_in[2];
    const float* Sa  = (const float*)d_in[3];
    const float* Wd1 = (const float*)d_in[4];  const float* bd1 = (const float*)d_in[5];
    const float* Wd2 = (const float*)d_in[6];  const float* bd2 = (const float*)d_in[7];
    const float* Wa1 = (const float*)d_in[8];  const float* ba1 = (const float*)d_in[9];
    const float* Wa2 = (const float*)d_in[10]; const float* ba2 = (const float*)d_in[11];
    const float* Wf  = (const float*)d_in[12]; const float* bf  = (const float*)d_in[13];

    float* ws = (float*)d_ws;
    float* x01  = ws;                       // 512*128
    float* T1d  = x01  + 65536;
    float* T2d  = T1d  + 65536;
    float* T1a  = T2d  + 65536;             // 8 * 512*128
    float* T2a  = T1a  + 8 * 65536;
    float* x02d = T2a  + 8 * 65536;         // 512*512
    float* x02a = x02d + 262144;
    float* U1d  = x02a + 262144;
    float* U2d  = U1d  + 262144;
    float* U1a  = U2d  + 262144;            // 8 * 512*512
    float* U2a  = U1a  + 8 * 262144;
    float* Gbuf = U2a  + 8 * 262144;        // 16 * 8192

    prep_x0<<<256, 256, 0, stream>>>(y, x01);

    // ---- layer 1 Chebyshev (F = 128): T1 = S@x0, T2 = 2*S@T1 - x0 ----
    cheb_mm<0><<<dim3(32, 1, 1), 256, 0, stream>>>(Sd, 0,      x01, 0,     x01,  0.f, T1d, 0,     128, 1.f);
    cheb_mm<0><<<dim3(32, 1, 8), 256, 0, stream>>>(Sa, 262144, x01, 0,     x01,  0.f, T1a, 65536, 128, 1.f);
    cheb_mm<1><<<dim3(32, 1, 1), 256, 0, stream>>>(Sd, 0,      T1d, 0,     x01, -1.f, T2d, 0,     128, 2.f);
    cheb_mm<1><<<dim3(32, 1, 8), 256, 0, stream>>>(Sa, 262144, T1a, 65536, x01, -1.f, T2a, 65536, 128, 2.f);

    // ---- layer 1 dense (tanh) -> layer-2 x0 layout ----
    dense_hidden<<<1024, 256, 0, stream>>>(x01, T1d, T2d, 0,     1, Wd1, bd1, x02d, 16, 64);
    dense_hidden<<<1024, 256, 0, stream>>>(x01, T1a, T2a, 65536, 8, Wa1, ba1, x02a, 16, 64);

    // ---- layer 2 Chebyshev (F = 512) ----
    cheb_mm<0><<<dim3(32, 4, 1), 256, 0, stream>>>(Sd, 0,      x02d, 0,      x02d,  0.f, U1d, 0,      512, 1.f);
    cheb_mm<0><<<dim3(32, 4, 8), 256, 0, stream>>>(Sa, 262144, x02a, 0,      x02a,  0.f, U1a, 262144, 512, 1.f);
    cheb_mm<1><<<dim3(32, 4, 1), 256, 0, stream>>>(Sd, 0,      U1d,  0,      x02d, -1.f, U2d, 0,      512, 2.f);
    cheb_mm<1><<<dim3(32, 4, 8), 256, 0, stream>>>(Sa, 262144, U1a,  262144, x02a, -1.f, U2a, 262144, 512, 2.f);

    // ---- layer 2 dense -> stacked gradients G (16 x 8192) ----
    dense_out<<<256, 256, 0, stream>>>(x02d, U1d, U2d, 0,      1, Wd2, bd2, Gbuf, 0, -0.1f);
    dense_out<<<256, 256, 0, stream>>>(x02a, U1a, U2a, 262144, 8, Wa2, ba2, Gbuf, 8, -1.f);

    // ---- fused gated GEMM over W_f (memory-bound stage, W_f read once) ----
    fused_gate_gemm<<<64, 256, 0, stream>>>(Gbuf, Wf, bf, (float*)d_out);
}